// QMixtralSparseMoeBlock_4475355922943
// MI455X (gfx1250) — compile-verified
//
#include <hip/hip_runtime.h>
#include <math.h>

typedef __attribute__((ext_vector_type(2))) float v2f;
typedef __attribute__((ext_vector_type(8))) float v8f;

constexpr int   TDIM = 4096;           // B*S tokens
constexpr int   HDIM = 2048;
constexpr int   FDIM = 7168;
constexpr int   EDIM = 8;
constexpr int   KC   = 32;             // K chunk staged in LDS
constexpr int   ASTR = 36;             // padded LDS row stride (floats): 16B aligned, bank-conflict free
constexpr float INV_GSCALE = 1.0f / (448.0f * 6.0f);   // 1/2688
constexpr float INV_FP8MAX = 1.0f / 448.0f;

#define WMMA_F32(A, B, C) \
  __builtin_amdgcn_wmma_f32_16x16x4_f32(false, (A), false, (B), (short)0, (C), false, false)

__device__ __forceinline__ void async_a_tile(unsigned lds_off, const float* gsrc) {
  asm volatile("global_load_async_to_lds_b128 %0, %1, off" :: "v"(lds_off), "v"(gsrc) : "memory");
}
__device__ __forceinline__ void wait_async0() {
  asm volatile("s_wait_asynccnt 0x0" ::: "memory");
}

// ---------------------------------------------------------------- router
__global__ __launch_bounds__(256)
void router_kernel(const float* __restrict__ x, const float* __restrict__ gw,
                   float* __restrict__ logits_out, float* __restrict__ w_tok,
                   float* __restrict__ rowmax_x) {
  const int t = blockIdx.x, tid = threadIdx.x;
  __shared__ float sm[256];
  __shared__ float lg[EDIM];
  float p[EDIM];
#pragma unroll
  for (int e = 0; e < EDIM; ++e) p[e] = 0.0f;
  float am = 0.0f;
  const float* xr = x + (size_t)t * HDIM;
  for (int i = tid; i < HDIM; i += 256) {
    const float xv = xr[i];
    am = fmaxf(am, fabsf(xv));
#pragma unroll
    for (int e = 0; e < EDIM; ++e) p[e] += xv * gw[e * HDIM + i];
  }
#pragma unroll
  for (int e = 0; e < EDIM; ++e) {
    sm[tid] = p[e]; __syncthreads();
    for (int s = 128; s > 0; s >>= 1) { if (tid < s) sm[tid] += sm[tid + s]; __syncthreads(); }
    if (tid == 0) lg[e] = sm[0];
    __syncthreads();
  }
  sm[tid] = am; __syncthreads();
  for (int s = 128; s > 0; s >>= 1) { if (tid < s) sm[tid] = fmaxf(sm[tid], sm[tid + s]); __syncthreads(); }
  if (tid == 0) {
    rowmax_x[t] = sm[0];
    float mx = lg[0];
    for (int e = 1; e < EDIM; ++e) mx = fmaxf(mx, lg[e]);
    float pr[EDIM];
    for (int e = 0; e < EDIM; ++e) pr[e] = __expf(lg[e] - mx);
    int i1 = 0;
    for (int e = 1; e < EDIM; ++e) if (pr[e] > pr[i1]) i1 = e;       // first-index ties like lax.top_k
    int i2 = (i1 == 0) ? 1 : 0;
    for (int e = 0; e < EDIM; ++e) if (e != i1 && pr[e] > pr[i2]) i2 = e;
    const float denom = pr[i1] + pr[i2];
    for (int e = 0; e < EDIM; ++e) {
      logits_out[(size_t)t * EDIM + e] = lg[e];
      w_tok[(size_t)t * EDIM + e] = (e == i1) ? pr[i1] / denom
                                  : (e == i2) ? pr[i2] / denom : 0.0f;
    }
  }
}

// -------------------------------------------------- deterministic compaction
__global__ void compact_kernel(const float* __restrict__ w_tok,
                               const float* __restrict__ rowmax_x,
                               int* __restrict__ token_list, int* __restrict__ counts,
                               float* __restrict__ gmax1,
                               int* __restrict__ tok2e, int* __restrict__ tok2s) {
  const int e = blockIdx.x, lane = threadIdx.x;   // blockDim = 32 (one wave)
  int cnt = 0;
  float gm = 0.0f;
  for (int t0 = 0; t0 < TDIM; t0 += 32) {
    const int t = t0 + lane;
    const bool sel = w_tok[(size_t)t * EDIM + e] > 0.0f;
    const unsigned bal = (unsigned)__ballot(sel);
    const int pre = __popc(bal & ((1u << lane) - 1u));
    if (sel) {
      const int slot = cnt + pre;
      token_list[(size_t)e * TDIM + slot] = t;
      int j = 0;
      for (int i = 0; i < e; ++i) j += (w_tok[(size_t)t * EDIM + i] > 0.0f) ? 1 : 0;
      tok2e[t * 2 + j] = e;
      tok2s[t * 2 + j] = slot;
      gm = fmaxf(gm, rowmax_x[t]);
    }
    cnt += __popc(bal);
  }
  for (int off = 16; off > 0; off >>= 1) gm = fmaxf(gm, __shfl_down(gm, off));
  if (lane == 0) { counts[e] = cnt; gmax1[e] = gm; }
}

// -------------------------------------------- offsets + outlier masks
__global__ void prep_kernel(const int* __restrict__ counts, int* __restrict__ offsets,
                            const int* __restrict__ r1, const int* __restrict__ r2,
                            const int* __restrict__ selp,
                            unsigned char* __restrict__ mask1, unsigned char* __restrict__ mask2) {
  const int tid = threadIdx.x;
  const int sel = selp[0];
  for (int idx = tid; idx < EDIM * sel; idx += 256) {
    const int e = idx / sel, i = idx - e * sel;
    mask1[(size_t)e * HDIM + r1[(size_t)e * HDIM + i]] = 1;
    mask2[(size_t)e * FDIM + r2[(size_t)e * FDIM + i]] = 1;
  }
  if (tid == 0) {
    int off = 0;
    for (int e = 0; e < EDIM; ++e) { offsets[e] = off; off += counts[e]; }
  }
}

// -------------------------------------------------------- quantize X
__global__ __launch_bounds__(256)
void quantx_kernel(const float* __restrict__ x, const int* __restrict__ counts,
                   const int* __restrict__ offsets, const int* __restrict__ token_list,
                   const float* __restrict__ gmax1, const float* __restrict__ rowmax_x,
                   const unsigned char* __restrict__ mask1, float* __restrict__ xq) {
  const int e = blockIdx.y, slot = blockIdx.x;
  if (slot >= counts[e]) return;
  const int t = token_list[(size_t)e * TDIM + slot];
  const float g = fmaxf(gmax1[e], 1e-8f) * INV_GSCALE;
  const float r = fmaxf(rowmax_x[t] / g, 1e-8f) * INV_FP8MAX;
  const float s = g * r, inv = 1.0f / s;
  const float* xr = x + (size_t)t * HDIM;
  float* o = xq + (size_t)(offsets[e] + slot) * HDIM;
  const unsigned char* mk = mask1 + (size_t)e * HDIM;
  for (int c = threadIdx.x; c < HDIM; c += 256) {
    const float xv = xr[c];
    const float q = fminf(fmaxf(rintf(xv * inv), -448.0f), 448.0f);
    o[c] = mk[c] ? xv : q * s;
  }
}

// ------------------------------------------ GEMM1: h = silu(Xq W1^T) * (Xq W3^T)
// Block tile 32(M) x 128(N); 8 waves, each wave: 2 M-subtiles x 16 N (B frag reused),
// A tile staged in LDS via async-to-LDS DMA, double buffered.
__global__ __launch_bounds__(256)
void gemm1_kernel(const float* __restrict__ xq, const float* __restrict__ w1,
                  const float* __restrict__ w3, float* __restrict__ hbuf,
                  const int* __restrict__ counts, const int* __restrict__ offsets,
                  unsigned* __restrict__ rowmaxh, unsigned* __restrict__ gmaxh) {
  __shared__ float As[2 * 32 * ASTR];
  const int e = blockIdx.z;
  const int cnt = counts[e];
  const int m0 = blockIdx.y * 32;
  if (m0 >= cnt) return;                      // uniform -> EXEC all ones below
  const int tid = threadIdx.x;
  const int wave = tid >> 5, lane = tid & 31;
  const int n0 = blockIdx.x * 128 + wave * 16;
  const int base = offsets[e];
  // A loader: thread tid -> row tid>>3 (0..31), 16B column chunk (tid&7)*4
  const int lrow = tid >> 3, lcol = (tid & 7) * 4;
  const int grow = min(m0 + lrow, cnt - 1);
  const float* agp = xq + (size_t)(base + grow) * HDIM + lcol;
  const unsigned lb0 = (unsigned)(uintptr_t)&As[lrow * ASTR + lcol];
  const unsigned lb1 = (unsigned)(uintptr_t)&As[32 * ASTR + lrow * ASTR + lcol];
  // fragment mapping
  const int mlane = lane & 15, koff = (lane >> 4) * 2;
  const float* __restrict__ B1 = w1 + ((size_t)e * FDIM + n0 + mlane) * HDIM + koff;
  const float* __restrict__ B3 = w3 + ((size_t)e * FDIM + n0 + mlane) * HDIM + koff;
  v8f acc1a = {}, acc1b = {}, acc3a = {}, acc3b = {};

  async_a_tile(lb0, agp);                     // prologue: chunk 0 -> buffer 0
  wait_async0();
  __syncthreads();
  for (int kc = 0; kc < HDIM; kc += KC) {
    const int bufsel = (kc / KC) & 1;
    if (kc + KC < HDIM)                       // prefetch next chunk -> other buffer
      async_a_tile(bufsel ? lb0 : lb1, agp + kc + KC);
    const float* Ab  = &As[bufsel * (32 * ASTR)];
    const float* a0p = Ab + mlane * ASTR + koff;
    const float* a1p = Ab + (mlane + 16) * ASTR + koff;
#pragma unroll
    for (int kk = 0; kk < KC; kk += 4) {
      v2f a0, a1, b1, b3;
      a0.x = a0p[kk];      a0.y = a0p[kk + 1];
      a1.x = a1p[kk];      a1.y = a1p[kk + 1];
      b1.x = B1[kc + kk];  b1.y = B1[kc + kk + 1];
      b3.x = B3[kc + kk];  b3.y = B3[kc + kk + 1];
      acc1a = WMMA_F32(a0, b1, acc1a);
      acc1b = WMMA_F32(a1, b1, acc1b);
      acc3a = WMMA_F32(a0, b3, acc3a);
      acc3b = WMMA_F32(a1, b3, acc3b);
    }
    wait_async0();
    __syncthreads();
  }
  const int nc = n0 + mlane;
  float rmax = 0.0f;
#pragma unroll
  for (int ms = 0; ms < 2; ++ms) {
    const v8f y1v = ms ? acc1b : acc1a;
    const v8f y3v = ms ? acc3b : acc3a;
    const int mbase = m0 + ms * 16 + ((lane >> 4) << 3);
#pragma unroll
    for (int v = 0; v < 8; ++v) {
      const int m = mbase + v;
      if (m < cnt) {
        const float y1 = y1v[v], y3 = y3v[v];
        const float hv = y1 * __builtin_amdgcn_rcpf(1.0f + __expf(-y1)) * y3;
        hbuf[(size_t)(base + m) * FDIM + nc] = hv;
        atomicMax(&rowmaxh[base + m], __float_as_uint(fabsf(hv)));
        rmax = fmaxf(rmax, fabsf(hv));
      }
    }
  }
  atomicMax(&gmaxh[e], __float_as_uint(rmax));
}

// -------------------------------------------------------- quantize H (in place)
__global__ __launch_bounds__(256)
void quanth_kernel(float* __restrict__ hbuf, const int* __restrict__ counts,
                   const int* __restrict__ offsets, const unsigned* __restrict__ rowmaxh,
                   const unsigned* __restrict__ gmaxh,
                   const unsigned char* __restrict__ mask2) {
  const int e = blockIdx.y, slot = blockIdx.x;
  if (slot >= counts[e]) return;
  const int row = offsets[e] + slot;
  const float g = fmaxf(__uint_as_float(gmaxh[e]), 1e-8f) * INV_GSCALE;
  const float r = fmaxf(__uint_as_float(rowmaxh[row]) / g, 1e-8f) * INV_FP8MAX;
  const float s = g * r, inv = 1.0f / s;
  float* o = hbuf + (size_t)row * FDIM;
  const unsigned char* mk = mask2 + (size_t)e * FDIM;
  for (int c = threadIdx.x; c < FDIM; c += 256) {
    const float hv = o[c];
    const float q = fminf(fmaxf(rintf(hv * inv), -448.0f), 448.0f);
    o[c] = mk[c] ? hv : q * s;
  }
}

// ------------------------------------------ GEMM2: Y = Hq W2^T (per slot)
__global__ __launch_bounds__(256)
void gemm2_kernel(const float* __restrict__ hq, const float* __restrict__ w2,
                  float* __restrict__ ybuf, const int* __restrict__ counts,
                  const int* __restrict__ offsets) {
  __shared__ float As[2 * 32 * ASTR];
  const int e = blockIdx.z;
  const int cnt = counts[e];
  const int m0 = blockIdx.y * 32;
  if (m0 >= cnt) return;
  const int tid = threadIdx.x;
  const int wave = tid >> 5, lane = tid & 31;
  const int n0 = blockIdx.x * 128 + wave * 16;
  const int base = offsets[e];
  const int lrow = tid >> 3, lcol = (tid & 7) * 4;
  const int grow = min(m0 + lrow, cnt - 1);
  const float* agp = hq + (size_t)(base + grow) * FDIM + lcol;
  const unsigned lb0 = (unsigned)(uintptr_t)&As[lrow * ASTR + lcol];
  const unsigned lb1 = (unsigned)(uintptr_t)&As[32 * ASTR + lrow * ASTR + lcol];
  const int mlane = lane & 15, koff = (lane >> 4) * 2;
  const float* __restrict__ Bw = w2 + ((size_t)e * HDIM + n0 + mlane) * FDIM + koff;
  v8f acca = {}, accb = {};

  async_a_tile(lb0, agp);
  wait_async0();
  __syncthreads();
  for (int kc = 0; kc < FDIM; kc += KC) {
    const int bufsel = (kc / KC) & 1;
    if (kc + KC < FDIM)
      async_a_tile(bufsel ? lb0 : lb1, agp + kc + KC);
    const float* Ab  = &As[bufsel * (32 * ASTR)];
    const float* a0p = Ab + mlane * ASTR + koff;
    const float* a1p = Ab + (mlane + 16) * ASTR + koff;
#pragma unroll
    for (int kk = 0; kk < KC; kk += 4) {
      v2f a0, a1, b;
      a0.x = a0p[kk];     a0.y = a0p[kk + 1];
      a1.x = a1p[kk];     a1.y = a1p[kk + 1];
      b.x  = Bw[kc + kk]; b.y  = Bw[kc + kk + 1];
      acca = WMMA_F32(a0, b, acca);
      accb = WMMA_F32(a1, b, accb);
    }
    wait_async0();
    __syncthreads();
  }
  const int nc = n0 + mlane;
#pragma unroll
  for (int ms = 0; ms < 2; ++ms) {
    const v8f acc = ms ? accb : acca;
    const int mbase = m0 + ms * 16 + ((lane >> 4) << 3);
#pragma unroll
    for (int v = 0; v < 8; ++v) {
      const int m = mbase + v;
      if (m < cnt) ybuf[(size_t)(base + m) * HDIM + nc] = acc[v];
    }
  }
}

// ----------------------------------------- combine two expert outputs per token
__global__ __launch_bounds__(256)
void combine_kernel(const float* __restrict__ ybuf, const float* __restrict__ w_tok,
                    const int* __restrict__ tok2e, const int* __restrict__ tok2s,
                    const int* __restrict__ offsets, float* __restrict__ out) {
  const int t = blockIdx.x, tid = threadIdx.x;
  const int e0 = tok2e[t * 2], e1 = tok2e[t * 2 + 1];
  const int r0 = offsets[e0] + tok2s[t * 2];
  const int r1 = offsets[e1] + tok2s[t * 2 + 1];
  const float wt0 = w_tok[(size_t)t * EDIM + e0];
  const float wt1 = w_tok[(size_t)t * EDIM + e1];
  const float* y0 = ybuf + (size_t)r0 * HDIM;
  const float* y1 = ybuf + (size_t)r1 * HDIM;
  float* o = out + (size_t)t * HDIM;
  for (int c = tid; c < HDIM; c += 256) o[c] = y0[c] * wt0 + y1[c] * wt1;
}

// ------------------------------------------------------------------ launch
extern "C" void kernel_launch(void* const* d_in, const int* in_sizes, int n_in,
                              void* d_out, int out_size, void* d_ws, size_t ws_size,
                              hipStream_t stream) {
  const float* x   = (const float*)d_in[0];
  const float* gw  = (const float*)d_in[1];
  const float* w1  = (const float*)d_in[2];
  const float* w3  = (const float*)d_in[3];
  const float* w2  = (const float*)d_in[4];
  const int*   r1  = (const int*)d_in[5];
  const int*   r2  = (const int*)d_in[6];
  const int*   sel = (const int*)d_in[7];

  float* out    = (float*)d_out;                       // [T, H]
  float* logits = out + (size_t)TDIM * HDIM;           // [T, E]

  size_t off = 0;
  auto alloc = [&](size_t bytes, size_t align) -> void* {
    off = (off + align - 1) & ~(align - 1);
    void* p = (char*)d_ws + off;
    off += bytes;
    return p;
  };
  float*         w_tok    = (float*)alloc((size_t)TDIM * EDIM * 4, 16);
  float*         rowmax_x = (float*)alloc((size_t)TDIM * 4, 16);
  float*         gmax1    = (float*)alloc(EDIM * 4, 16);
  int*           counts   = (int*)alloc(EDIM * 4, 16);
  int*           offsets  = (int*)alloc(EDIM * 4, 16);
  int*           tlist    = (int*)alloc((size_t)EDIM * TDIM * 4, 16);
  int*           tok2e    = (int*)alloc((size_t)2 * TDIM * 4, 16);
  int*           tok2s    = (int*)alloc((size_t)2 * TDIM * 4, 16);
  // zero-initialized region (single memset): gmaxh, rowmaxh, mask1, mask2
  unsigned*      gmaxh    = (unsigned*)alloc(EDIM * 4, 16);
  unsigned*      rowmaxh  = (unsigned*)alloc((size_t)2 * TDIM * 4, 16);
  unsigned char* mask1    = (unsigned char*)alloc((size_t)EDIM * HDIM, 16);
  unsigned char* mask2    = (unsigned char*)alloc((size_t)EDIM * FDIM, 16);
  float*         ybuf     = (float*)alloc((size_t)2 * TDIM * HDIM * 4, 256); // xq then reused as Y
  float*         hbuf     = (float*)alloc((size_t)2 * TDIM * FDIM * 4, 256);
  float*         xq       = ybuf;

  const size_t zbytes = (size_t)((char*)mask2 + (size_t)EDIM * FDIM - (char*)gmaxh);
  hipMemsetAsync(gmaxh, 0, zbytes, stream);

  router_kernel<<<TDIM, 256, 0, stream>>>(x, gw, logits, w_tok, rowmax_x);
  compact_kernel<<<EDIM, 32, 0, stream>>>(w_tok, rowmax_x, tlist, counts, gmax1, tok2e, tok2s);
  prep_kernel<<<1, 256, 0, stream>>>(counts, offsets, r1, r2, sel, mask1, mask2);
  quantx_kernel<<<dim3(TDIM, EDIM), 256, 0, stream>>>(x, counts, offsets, tlist, gmax1,
                                                      rowmax_x, mask1, xq);
  gemm1_kernel<<<dim3(FDIM / 128, TDIM / 32, EDIM), 256, 0, stream>>>(xq, w1, w3, hbuf,
                                                                      counts, offsets,
                                                                      rowmaxh, gmaxh);
  quanth_kernel<<<dim3(TDIM, EDIM), 256, 0, stream>>>(hbuf, counts, offsets, rowmaxh,
                                                      gmaxh, mask2);
  gemm2_kernel<<<dim3(HDIM / 128, TDIM / 32, EDIM), 256, 0, stream>>>(hbuf, w2, ybuf,
                                                                      counts, offsets);
  combine_kernel<<<TDIM, 256, 0, stream>>>(ybuf, w_tok, tok2e, tok2s, offsets, out);
}